// GCN_4252017623589
// MI455X (gfx1250) — compile-verified
//
#include <hip/hip_runtime.h>

typedef __attribute__((ext_vector_type(2))) float v2f;
typedef __attribute__((ext_vector_type(8))) float v8f;

#define NGRAPH 512

// ---------------- utility kernels ----------------

__global__ void zero_f32(float* __restrict__ p, int n) {
  int i = blockIdx.x * blockDim.x + threadIdx.x;
  if (i < n) p[i] = 0.0f;
}

__global__ void deg_edges(const int* __restrict__ dst, float* __restrict__ deg, int E) {
  int e = blockIdx.x * blockDim.x + threadIdx.x;
  if (e < E) atomicAdd(&deg[dst[e]], 1.0f);
}

// deg -> deg^-1/2 in place (self-loop adds +1, so deg >= 1 always)
__global__ void make_dinv(float* __restrict__ deg, int n) {
  int i = blockIdx.x * blockDim.x + threadIdx.x;
  if (i < n) deg[i] = rsqrtf(deg[i] + 1.0f);
}

// Layer-1 GEMM: [N,3] @ [3,32]; K=3 too skinny for WMMA.
// Fused epilogue: H = xW, Init = H * dinv^2 (self-loop term of the aggregation).
__global__ void gemm_k3_self(const float* __restrict__ x, const float* __restrict__ W,
                             const float* __restrict__ dinv,
                             float* __restrict__ H, float* __restrict__ Init, int n) {
  int i = blockIdx.x * blockDim.x + threadIdx.x;
  if (i >= n * 32) return;
  int r = i >> 5, c = i & 31;
  const float* xr = x + (size_t)r * 3;
  float v = fmaf(xr[0], W[c], fmaf(xr[1], W[32 + c], xr[2] * W[64 + c]));
  H[i] = v;
  float w = dinv[r];
  Init[i] = v * w * w;
}

// ---------------- f32 WMMA GEMM, column-tile-per-wave ----------------
// Out[N, NCOLS] = A[N, K] @ W[K, NCOLS] via V_WMMA_F32_16X16X4_F32.
// Each wave owns ONE 16-column tile and MT consecutive 16-row tiles; all B
// fragments for its column are hoisted into registers once (K/2 floats/lane),
// so the inner loop is one global_load_b64 (A frag) per WMMA.
// If SELF: also write Init = Out * dinv[row]^2 (fused self-loop init).
template <int K, int NCOLS, int MT, bool SELF>
__global__ void wmma_gemm_f32(const float* __restrict__ A, const float* __restrict__ W,
                              float* __restrict__ Out, float* __restrict__ Init,
                              const float* __restrict__ dinv, int nrows) {
  constexpr int NT = NCOLS / 16;
  const int lane = threadIdx.x & 31;
  const int wave = threadIdx.x >> 5;
  const int wid  = blockIdx.x * (blockDim.x >> 5) + wave;
  const int t    = wid % NT;            // column tile
  const int rb   = wid / NT;            // row block (MT row tiles)
  const int base = rb * 16 * MT;
  if (base >= nrows) return;            // wave-uniform: EXEC stays all-ones
  const int half = lane >> 4;           // K-half selector
  const int l16  = lane & 15;
  const int col  = t * 16 + l16;

  // Hoist B fragments for this column tile (reused across MT row tiles).
  v2f bfr[K / 4];
#pragma unroll
  for (int kk = 0; kk < K / 4; ++kk) {
    const float* bp = W + (size_t)(kk * 4 + half * 2) * NCOLS + col;
    bfr[kk].x = bp[0];
    bfr[kk].y = bp[NCOLS];
  }

#pragma unroll
  for (int m = 0; m < MT; ++m) {
    const int row0 = base + m * 16;
    if (row0 >= nrows) break;           // wave-uniform
    const float* arow = A + (size_t)(row0 + l16) * K + half * 2;
    v8f acc = (v8f){0.f,0.f,0.f,0.f,0.f,0.f,0.f,0.f};
#pragma unroll
    for (int kk = 0; kk < K / 4; ++kk) {
      v2f afrag;                        // contiguous pair -> one global_load_b64
      afrag.x = arow[kk * 4 + 0];
      afrag.y = arow[kk * 4 + 1];
      acc = __builtin_amdgcn_wmma_f32_16x16x4_f32(
          false, afrag, false, bfr[kk], (short)0, acc, false, false);
    }
    // C/D layout: VGPR r -> row (row0 + half*8 + r), this wave's column tile
#pragma unroll
    for (int r = 0; r < 8; ++r) {
      int row = row0 + half * 8 + r;
      float v = acc[r];
      Out[(size_t)row * NCOLS + col] = v;
      if (SELF) {
        float w = dinv[row];
        Init[(size_t)row * NCOLS + col] = v * w * w;
      }
    }
  }
}

// ---------------- aggregation (L2-resident scatter) ----------------

// out[dst] += H[src] * dinv[src]*dinv[dst], one thread per (edge, 4-col chunk)
__global__ void agg_edges(const float* __restrict__ H, const float* __restrict__ dinv,
                          const int* __restrict__ src, const int* __restrict__ dst,
                          float* __restrict__ out, int E, int dim4) {
  int i = blockIdx.x * blockDim.x + threadIdx.x;
  if (i >= E * dim4) return;
  int e = i / dim4, c4 = i % dim4;
  int s = src[e], d = dst[e];
  float w = dinv[s] * dinv[d];
  float4 v = ((const float4*)H)[(size_t)s * dim4 + c4];
  float* o = out + ((size_t)d * dim4 + c4) * 4;
  atomicAdd(o + 0, v.x * w);
  atomicAdd(o + 1, v.y * w);
  atomicAdd(o + 2, v.z * w);
  atomicAdd(o + 3, v.w * w);
}

__global__ void bias_relu(float* __restrict__ h, const float* __restrict__ b,
                          int n, int dim) {
  int i = blockIdx.x * blockDim.x + threadIdx.x;
  if (i >= n * dim) return;
  float v = h[i] + b[i % dim];
  h[i] = v > 0.0f ? v : 0.0f;
}

// ---------------- pooling + head ----------------

__global__ void pool_accum(const float* __restrict__ H, const int* __restrict__ batch,
                           float* __restrict__ pooled, float* __restrict__ cnt, int n) {
  int i = blockIdx.x * blockDim.x + threadIdx.x;   // n * 32 (dim4 = 32 for dim=128)
  if (i >= n * 32) return;
  int r = i >> 5, c4 = i & 31;
  int g = batch[r];
  float4 v = ((const float4*)H)[i];
  float* p = pooled + ((size_t)g * 32 + c4) * 4;
  atomicAdd(p + 0, v.x);
  atomicAdd(p + 1, v.y);
  atomicAdd(p + 2, v.z);
  atomicAdd(p + 3, v.w);
  if (c4 == 0) atomicAdd(&cnt[g], 1.0f);
}

__global__ void pool_div(float* __restrict__ pooled, const float* __restrict__ cnt) {
  int i = blockIdx.x * blockDim.x + threadIdx.x;   // NGRAPH * 128
  if (i >= NGRAPH * 128) return;
  float c = cnt[i >> 7];
  pooled[i] /= fmaxf(c, 1.0f);
}

__global__ void fc2(const float* __restrict__ z, const float* __restrict__ W,
                    const float* __restrict__ b, float* __restrict__ out) {
  int i = blockIdx.x * blockDim.x + threadIdx.x;   // NGRAPH * 4
  if (i >= NGRAPH * 4) return;
  int g = i >> 2, c = i & 3;
  const float* zr = z + (size_t)g * 32;
  float acc = b[c];
#pragma unroll
  for (int k = 0; k < 32; ++k) acc = fmaf(zr[k], W[k * 4 + c], acc);
  out[i] = acc;
}

// ---------------- launch ----------------

static inline int cdiv(long long a, int b) { return (int)((a + b - 1) / b); }

extern "C" void kernel_launch(void* const* d_in, const int* in_sizes, int n_in,
                              void* d_out, int out_size, void* d_ws, size_t ws_size,
                              hipStream_t stream) {
  const float* x   = (const float*)d_in[0];
  const int*   ei  = (const int*)d_in[1];
  const int*   bat = (const int*)d_in[2];
  const float* W1  = (const float*)d_in[3];
  const float* b1  = (const float*)d_in[4];
  const float* W2  = (const float*)d_in[5];
  const float* b2  = (const float*)d_in[6];
  const float* W3  = (const float*)d_in[7];
  const float* b3  = (const float*)d_in[8];
  const float* Wf1 = (const float*)d_in[9];
  const float* bf1 = (const float*)d_in[10];
  const float* Wf2 = (const float*)d_in[11];
  const float* bf2 = (const float*)d_in[12];
  float* out = (float*)d_out;

  const int N = in_sizes[0] / 3;     // 50000
  const int E = in_sizes[1] / 2;     // 800000
  const int* src = ei;               // edge_index[0]
  const int* dst = ei + E;           // edge_index[1]

  // workspace (floats): dinv[N] | Hbuf[N*128] | act0[N*128] | act1[N*128]
  //                     | pooled[512*128] | cnt[512] | z1[512*32]
  float* ws     = (float*)d_ws;
  float* dinv   = ws;
  float* Hbuf   = dinv + N;
  float* act0   = Hbuf + (size_t)N * 128;
  float* act1   = act0 + (size_t)N * 128;
  float* pooled = act1 + (size_t)N * 128;
  float* cnt    = pooled + NGRAPH * 128;
  float* z1     = cnt + NGRAPH;

  const int BT = 256;  // 8 wave32 waves per block
  const int WPB = BT / 32;

  // degree -> dinv
  zero_f32 <<<cdiv(N, BT), BT, 0, stream>>>(dinv, N);
  deg_edges<<<cdiv(E, BT), BT, 0, stream>>>(dst, dinv, E);
  make_dinv<<<cdiv(N, BT), BT, 0, stream>>>(dinv, N);

  // ---- layer 1: x[N,3] @ W1[3,32] -> Hbuf, self-loop init -> act1 ----
  gemm_k3_self<<<cdiv((long long)N * 32, BT), BT, 0, stream>>>(x, W1, dinv, Hbuf, act1, N);
  agg_edges<<<cdiv((long long)E * 8, BT), BT, 0, stream>>>(Hbuf, dinv, src, dst, act1, E, 8);
  bias_relu<<<cdiv((long long)N * 32, BT), BT, 0, stream>>>(act1, b1, N, 32);

  // ---- layer 2: act1[N,32] @ W2[32,64] -> Hbuf, init -> act0 ----
  {
    constexpr int MT = 5;                         // 3125 row tiles = 625 * 5
    int waves = cdiv(N, 16 * MT) * (64 / 16);
    wmma_gemm_f32<32, 64, MT, true>
        <<<cdiv(waves, WPB), BT, 0, stream>>>(act1, W2, Hbuf, act0, dinv, N);
  }
  agg_edges<<<cdiv((long long)E * 16, BT), BT, 0, stream>>>(Hbuf, dinv, src, dst, act0, E, 16);
  bias_relu<<<cdiv((long long)N * 64, BT), BT, 0, stream>>>(act0, b2, N, 64);

  // ---- layer 3: act0[N,64] @ W3[64,128] -> Hbuf, init -> act1 ----
  {
    constexpr int MT = 5;
    int waves = cdiv(N, 16 * MT) * (128 / 16);
    wmma_gemm_f32<64, 128, MT, true>
        <<<cdiv(waves, WPB), BT, 0, stream>>>(act0, W3, Hbuf, act1, dinv, N);
  }
  agg_edges<<<cdiv((long long)E * 32,  BT), BT, 0, stream>>>(Hbuf, dinv, src, dst, act1, E, 32);
  bias_relu<<<cdiv((long long)N * 128, BT), BT, 0, stream>>>(act1, b3, N, 128);

  // ---- global mean pool ----
  zero_f32  <<<cdiv(NGRAPH * 128 + NGRAPH, BT), BT, 0, stream>>>(pooled, NGRAPH * 128 + NGRAPH);
  pool_accum<<<cdiv((long long)N * 32, BT), BT, 0, stream>>>(act1, bat, pooled, cnt, N);
  pool_div  <<<cdiv(NGRAPH * 128, BT), BT, 0, stream>>>(pooled, cnt);

  // ---- head: fc1 (WMMA, K=128) -> relu -> fc2 ----
  {
    constexpr int MT = 2;                         // 32 row tiles = 16 * 2
    int waves = cdiv(NGRAPH, 16 * MT) * (32 / 16);
    wmma_gemm_f32<128, 32, MT, false>
        <<<cdiv(waves, WPB), BT, 0, stream>>>(pooled, Wf1, z1, nullptr, nullptr, NGRAPH);
  }
  bias_relu<<<cdiv(NGRAPH * 32, BT), BT, 0, stream>>>(z1, bf1, NGRAPH, 32);
  fc2      <<<cdiv(NGRAPH * 4,  BT), BT, 0, stream>>>(z1, Wf2, bf2, out);
}